// GilbertMask_70205535420562
// MI455X (gfx1250) — compile-verified
//
#include <hip/hip_runtime.h>
#include <cstdint>

// ---------------- problem geometry ----------------
#define B_DIM 16
#define T_DIM 4096
#define C_DIM 1024
#define NROWS (B_DIM * T_DIM)                  // 65536 (b,t) rows
#define OUT_ELEMS ((long long)NROWS * C_DIM)   // 67108864 floats of `out`
#define U_HALF 32760u                          // ((T-1)*B)/2 for threefry counter split
#define ROWS_PER_BLOCK 4

typedef float v4f __attribute__((ext_vector_type(4)));
typedef int   v4i __attribute__((ext_vector_type(4)));

// ---------------- JAX Threefry-2x32 (bit-exact) ----------------
__host__ __device__ inline uint32_t rotl32(uint32_t x, uint32_t r) {
  return (x << r) | (x >> (32u - r));
}

__host__ __device__ inline void threefry2x32(uint32_t k0, uint32_t k1,
                                             uint32_t x0, uint32_t x1,
                                             uint32_t& o0, uint32_t& o1) {
  const uint32_t ks2 = k0 ^ k1 ^ 0x1BD11BDAu;
  uint32_t v0 = x0 + k0, v1 = x1 + k1;
#define TF_R(rot) { v0 += v1; v1 = rotl32(v1, rot); v1 ^= v0; }
  TF_R(13) TF_R(15) TF_R(26) TF_R(6)  v0 += k1;  v1 += ks2 + 1u;
  TF_R(17) TF_R(29) TF_R(16) TF_R(24) v0 += ks2; v1 += k0 + 2u;
  TF_R(13) TF_R(15) TF_R(26) TF_R(6)  v0 += k0;  v1 += k1 + 3u;
  TF_R(17) TF_R(29) TF_R(16) TF_R(24) v0 += k1;  v1 += ks2 + 4u;
  TF_R(13) TF_R(15) TF_R(26) TF_R(6)  v0 += ks2; v1 += k0 + 5u;
#undef TF_R
  o0 = v0; o1 = v1;
}

// element `idx` of jax.random.uniform(key, (n,)) with n = 2*half (even):
// counters are iota(n) split into halves (j, j+half).
__device__ inline float jax_uniform_elem(uint32_t k0, uint32_t k1,
                                         uint32_t idx, uint32_t half) {
  uint32_t y0, y1, bits;
  if (idx < half) { threefry2x32(k0, k1, idx, idx + half, y0, y1); bits = y0; }
  else            { threefry2x32(k0, k1, idx - half, idx, y0, y1); bits = y1; }
  return __uint_as_float((bits >> 9) | 0x3f800000u) - 1.0f;
}

// ---------------- kernel 1: Gilbert-Elliott mask via wave-parallel scan ----
// One block (= one wave32) per batch row. Each lane owns 128 time steps:
// builds its chunk's {0,1}->{0,1} transfer map, Kogge-Stone composes maps
// across the wave, then replays its chunk from the scanned entry state.
__global__ __launch_bounds__(32)
void gilbert_scan_kernel(float* __restrict__ maskOut,
                         uint32_t ki0, uint32_t ki1,   // key for init uniforms
                         uint32_t ko0, uint32_t ko1,   // key for u_from_one
                         uint32_t kz0, uint32_t kz1) { // key for u_from_zero
  const int b    = blockIdx.x;
  const int lane = threadIdx.x;          // 0..31
  const int CH   = T_DIM / 32;           // 128 steps per lane

  const float P_STAY1 = (float)(1.0 - 0.3 / (10.0 * 0.7)); // uo <= p -> stay 1
  const float P_01    = 0.1f;                               // uz <= p -> go 1
  const float P_INIT  = 0.3f;                               // u > p -> start 1

  uint32_t aw[4], bw[4];
  int A = 1, Bs = 0;                     // chunk map: entry1->A, entry0->Bs
#pragma unroll
  for (int w = 0; w < 4; ++w) {
    uint32_t av = 0, bv = 0;
    const int jjStart = (w == 0 && lane == 0) ? 1 : 0;  // t=0 is the init, not a step
    for (int jj = jjStart; jj < 32; ++jj) {
      const int j = w * 32 + jj;
      const int t = lane * CH + j;
      const uint32_t idx = (uint32_t)(t - 1) * 16u + (uint32_t)b; // (t-1, b) flat
      const float uo = jax_uniform_elem(ko0, ko1, idx, U_HALF);
      const float uz = jax_uniform_elem(kz0, kz1, idx, U_HALF);
      const int a  = (uo <= P_STAY1) ? 1 : 0;
      const int bb = (uz <= P_01)    ? 1 : 0;
      av |= (uint32_t)a  << jj;
      bv |= (uint32_t)bb << jj;
      A  = A  ? a : bb;
      Bs = Bs ? a : bb;
    }
    aw[w] = av; bw[w] = bv;
  }

  int initState = 0;
  int mymap;
  if (lane == 0) {
    // init = (uniform(k_init,(16,))[b] > 0.3); counters iota(16) split (b, b+8)
    uint32_t y0, y1, bits;
    if (b < 8) { threefry2x32(ki0, ki1, (uint32_t)b, (uint32_t)b + 8u, y0, y1); bits = y0; }
    else       { threefry2x32(ki0, ki1, (uint32_t)b - 8u, (uint32_t)b, y0, y1); bits = y1; }
    const float u0 = __uint_as_float((bits >> 9) | 0x3f800000u) - 1.0f;
    initState = (u0 > P_INIT) ? 1 : 0;
    const int sEnd = initState ? A : Bs;   // lane 0's map is constant
    mymap = (sEnd << 1) | sEnd;
  } else {
    mymap = (A << 1) | Bs;
  }

  // inclusive Kogge-Stone scan; compose(prev, cur): s -> cur(prev(s))
  int m = mymap;
#pragma unroll
  for (int d = 1; d < 32; d <<= 1) {
    const int pm = __shfl_up(m, d, 32);
    if (lane >= d) {
      const int na = ((pm >> 1) & 1) ? ((m >> 1) & 1) : (m & 1);
      const int nb = (pm & 1)        ? ((m >> 1) & 1) : (m & 1);
      m = (na << 1) | nb;
    }
  }
  const int em = __shfl_up(m, 1, 32);    // composed map of lanes 0..lane-1

  int s;
  const long long base = (long long)b * T_DIM + (long long)lane * CH;
  if (lane == 0) {
    s = initState;
    maskOut[base] = (float)s;            // t = 0
  } else {
    s = (em >> 1) & 1;                   // constant map (lane0 folded init in)
  }
  // replay chunk from packed decision bits
#pragma unroll
  for (int w = 0; w < 4; ++w) {
    const uint32_t av = aw[w], bv = bw[w];
    const int jjStart = (w == 0 && lane == 0) ? 1 : 0;
    for (int jj = jjStart; jj < 32; ++jj) {
      const int a  = (int)((av >> jj) & 1u);
      const int bb = (int)((bv >> jj) & 1u);
      s = s ? a : bb;
      maskOut[base + w * 32 + jj] = (float)s;
    }
  }
}

// ---------------- kernel 2: out = m*x + (1-m)*emb (HBM streaming) ----------
// CDNA5 path: async global->LDS (ASYNCcnt) double-buffered per 4KB row; each
// lane reads back exactly the 16B it staged (no cross-wave LDS dep, so no
// barrier needed — s_wait_asynccnt alone orders the DS read). Output stores
// are non-temporal (written once, never re-read; keep them out of L2).
#if defined(__HIP_DEVICE_COMPILE__) && \
    __has_builtin(__builtin_amdgcn_global_load_async_to_lds_b128) && \
    __has_builtin(__builtin_amdgcn_s_wait_asynccnt)
#define USE_ASYNC_LDS 1
#else
#define USE_ASYNC_LDS 0
#endif

// builtin signature (from clang diagnostic): param0 = v4i AS1* ("__device__"),
// param1 = v4i AS3* ("__shared__"), then imm offset, imm cpol.
typedef v4i __attribute__((address_space(1)))* as1_v4i;
typedef v4i __attribute__((address_space(3)))* as3_v4i;

__global__ __launch_bounds__(256)
void apply_mask_kernel(const float* __restrict__ x,
                       const float* __restrict__ emb,
                       const float* __restrict__ mask,
                       float* __restrict__ out) {
  const int tid = threadIdx.x;
  const long long row0 = (long long)blockIdx.x * ROWS_PER_BLOCK;
  const v4f e = *(const v4f*)(emb + tid * 4);   // 4KB, L2-resident, RT policy

#if USE_ASYNC_LDS
  __shared__ float buf[2][C_DIM];               // 2 x 4KB double buffer
  {
    float* g = const_cast<float*>(x) + row0 * C_DIM + (long long)tid * 4;
    __builtin_amdgcn_global_load_async_to_lds_b128(
        (as1_v4i)g, (as3_v4i)&buf[0][tid * 4], 0, 0);
  }
#pragma unroll
  for (int r = 0; r < ROWS_PER_BLOCK; ++r) {
    if (r + 1 < ROWS_PER_BLOCK) {
      float* g = const_cast<float*>(x) + (row0 + r + 1) * C_DIM + (long long)tid * 4;
      __builtin_amdgcn_global_load_async_to_lds_b128(
          (as1_v4i)g, (as3_v4i)&buf[(r + 1) & 1][tid * 4], 0, 0);
      __builtin_amdgcn_s_wait_asynccnt(1);      // previous row's stage done
    } else {
      __builtin_amdgcn_s_wait_asynccnt(0);
    }
    const v4f  xv = *(const v4f*)(&buf[r & 1][tid * 4]);   // ds_load_b128
    const float mv = mask[row0 + r];                       // uniform -> s_load
    const v4f  o  = xv * mv + e * (1.0f - mv);
    __builtin_nontemporal_store(o, (v4f*)(out + (row0 + r) * C_DIM + (long long)tid * 4));
  }
#else
#pragma unroll
  for (int r = 0; r < ROWS_PER_BLOCK; ++r) {
    const v4f  xv = __builtin_nontemporal_load(
        (const v4f*)(x + (row0 + r) * C_DIM + (long long)tid * 4));
    const float mv = mask[row0 + r];
    const v4f  o  = xv * mv + e * (1.0f - mv);
    __builtin_nontemporal_store(o, (v4f*)(out + (row0 + r) * C_DIM + (long long)tid * 4));
  }
#endif
}

// ---------------- host launcher ----------------
extern "C" void kernel_launch(void* const* d_in, const int* in_sizes, int n_in,
                              void* d_out, int out_size, void* d_ws, size_t ws_size,
                              hipStream_t stream) {
  (void)in_sizes; (void)n_in; (void)out_size; (void)d_ws; (void)ws_size;
  const float* x   = (const float*)d_in[0];
  const float* emb = (const float*)d_in[1];
  float* out     = (float*)d_out;
  float* maskOut = out + OUT_ELEMS;   // tuple: (out, mask) flat-concatenated

  // jax.random.split(jax.random.key(1), 3):
  // threefry2x32(key=(0,1), iota(6)) with counter halves (0,1,2)/(3,4,5),
  // output [y0_0,y0_1,y0_2,y1_0,y1_1,y1_2] reshaped to (3,2).
  uint32_t ya[3], yb[3];
  for (uint32_t i = 0; i < 3; ++i) threefry2x32(0u, 1u, i, i + 3u, ya[i], yb[i]);
  const uint32_t ki0 = ya[0], ki1 = ya[1];   // k0 -> init state uniforms
  const uint32_t ko0 = ya[2], ko1 = yb[0];   // k1 -> u_from_one
  const uint32_t kz0 = yb[1], kz1 = yb[2];   // k2 -> u_from_zero

  hipLaunchKernelGGL(gilbert_scan_kernel, dim3(B_DIM), dim3(32), 0, stream,
                     maskOut, ki0, ki1, ko0, ko1, kz0, kz1);
  hipLaunchKernelGGL(apply_mask_kernel, dim3(NROWS / ROWS_PER_BLOCK), dim3(256),
                     0, stream, x, emb, maskOut, out);
}